// STSProbSparseSelfAttention_27058293965112
// MI455X (gfx1250) — compile-verified
//
#include <hip/hip_runtime.h>
#include <hip/hip_bf16.h>

// ---------------- problem constants (from reference) ----------------
#define KN_   2048
#define B_    4
#define D_    512
#define H_    8
#define E_    64
#define U_    40            // U_PART = 5*ceil(ln(2048)) = 40
#define GM_   (KN_ * B_)    // 8192 GEMM rows
#define GN_   D_            // 512
#define GK_   D_            // 512

// GEMM tiling
#define BM_   128
#define BN_   64
#define BK_   16

typedef float v2f __attribute__((ext_vector_type(2)));
typedef float v8f __attribute__((ext_vector_type(8)));
typedef unsigned int u32x4 __attribute__((ext_vector_type(4)));
typedef int i32x4 __attribute__((ext_vector_type(4)));
typedef int i32x8 __attribute__((ext_vector_type(8)));

#define HAVE_TDM __has_builtin(__builtin_amdgcn_tensor_load_to_lds)

#if HAVE_TDM
// ---------------------------------------------------------------------
// Issue a TDM 2D tile load: global (row-major, stride_elems f32/row,
// tile_d0 f32 per row contiguous, tile_d1 rows) -> LDS (packed row-major).
// D# layout per CDNA5 ISA ch. 8 (groups 0/1; groups 2/3 zero for 2D).
// 6-arg builtin form (clang-23 / therock-10.0 headers).
// ---------------------------------------------------------------------
__device__ __forceinline__ void tdm_load_2d(unsigned lds_addr, const float* gptr,
                                            unsigned tile_d0, unsigned tile_d1,
                                            unsigned stride_elems)
{
    unsigned long long ga = (unsigned long long)(uintptr_t)gptr;
    u32x4 g0;
    g0.x = 1u;                                               // count=1, user desc
    g0.y = lds_addr;                                         // LDS byte address
    g0.z = (unsigned)ga;                                     // global_addr[31:0]
    g0.w = (unsigned)((ga >> 32) & 0x1FFFFFFu) | (2u << 30); // addr[56:32] | type=2
    i32x8 g1;
    g1[0] = (int)(2u << 16);                                  // data_size = 4 bytes
    g1[1] = (int)((tile_d0 & 0xFFFFu) << 16);                 // tensor_dim0[15:0]
    g1[2] = (int)((tile_d0 >> 16) | ((tile_d1 & 0xFFFFu) << 16)); // td0 hi | td1 lo
    g1[3] = (int)((tile_d1 >> 16) | (tile_d0 << 16));         // td1 hi | tile_dim0
    g1[4] = (int)(tile_d1 & 0xFFFFu);                         // tile_dim1 | tile_dim2=0
    g1[5] = (int)stride_elems;                                // dim0 stride low 32
    g1[6] = 0;                                                // stride hi | dim1 stride lo
    g1[7] = 0;
    i32x4 z4 = {0, 0, 0, 0};
    i32x8 z8 = {0, 0, 0, 0, 0, 0, 0, 0};
    __builtin_amdgcn_tensor_load_to_lds(g0, g1, z4, z4, z8, 0);
}
#endif

// =====================================================================
// WMMA f32 GEMM:  OUT = (X @ W^T + bias) * scale
//   X: (GM_, GK_) row-major ; W: (GN_, GK_) row-major (B = W^T)
// Double-buffered LDS tiles staged by the Tensor Data Mover (wave 0),
// all 8 waves run the V_WMMA_F32_16X16X4_F32 pipe.
// layout==0: X rows ordered (n*B + b); OUT stored [b][h][n][e]
// layout==1: OUT row-major (GM_, GN_)
// =====================================================================
__global__ __launch_bounds__(256)
void wmma_gemm_kernel(const float* __restrict__ X, const float* __restrict__ W,
                      const float* __restrict__ bias, float scale,
                      float* __restrict__ OUT, int layout)
{
    __shared__ float As[2][BM_ * BK_];   // [buf][row*16 + k]
    __shared__ float Bs[2][BN_ * BK_];   // [buf][col*16 + k]

    const int tid  = threadIdx.x;
    const int wave = tid >> 5;
    const int lane = tid & 31;
    const int g    = lane >> 4;          // K-pair half (A/B), M-half (C/D)
    const int l16  = lane & 15;

    const int tileM = blockIdx.y * BM_;
    const int tileN = blockIdx.x * BN_;

    const float* Abase = X + (size_t)tileM * GK_;
    const float* Bbase = W + (size_t)tileN * GK_;

    v8f acc[4];
#pragma unroll
    for (int i = 0; i < 4; ++i)
#pragma unroll
        for (int j = 0; j < 8; ++j) acc[i][j] = 0.0f;

    const int nIter = GK_ / BK_;         // 32

#if HAVE_TDM
    // LDS byte offsets (generic-pointer low 32 bits == LDS offset per aperture)
    const unsigned AsAddr0 = (unsigned)(uintptr_t)(&As[0][0]);
    const unsigned AsAddr1 = (unsigned)(uintptr_t)(&As[1][0]);
    const unsigned BsAddr0 = (unsigned)(uintptr_t)(&Bs[0][0]);
    const unsigned BsAddr1 = (unsigned)(uintptr_t)(&Bs[1][0]);

    if (wave == 0) {                     // prologue: stage buffer 0
        tdm_load_2d(AsAddr0, Abase, BK_, BM_, GK_);
        tdm_load_2d(BsAddr0, Bbase, BK_, BN_, GK_);
    }
    for (int i = 0; i < nIter; ++i) {
        const int cur = i & 1;
        if (wave == 0) {
            if (i + 1 < nIter) {         // stage next buffer, then wait current pair
                const unsigned aN = cur ? AsAddr0 : AsAddr1;
                const unsigned bN = cur ? BsAddr0 : BsAddr1;
                tdm_load_2d(aN, Abase + (i + 1) * BK_, BK_, BM_, GK_);
                tdm_load_2d(bN, Bbase + (i + 1) * BK_, BK_, BN_, GK_);
                __builtin_amdgcn_s_wait_tensorcnt(2);  // in-order: current pair done
            } else {
                __builtin_amdgcn_s_wait_tensorcnt(0);
            }
        }
        __syncthreads();                 // release compute on buffer `cur`
#pragma unroll
        for (int kk = 0; kk < BK_; kk += 4) {
            // A fragment (16x4 f32): lanes 0-15 K=kk..kk+1, lanes 16-31 K=kk+2..kk+3
            v2f a = *(const v2f*)(&As[cur][(wave * 16 + l16) * BK_ + kk + 2 * g]);
#pragma unroll
            for (int nt = 0; nt < 4; ++nt) {
                // B fragment: (W[col][k], W[col][k+1]) contiguous in the TDM tile
                v2f b = *(const v2f*)(&Bs[cur][(nt * 16 + l16) * BK_ + kk + 2 * g]);
                acc[nt] = __builtin_amdgcn_wmma_f32_16x16x4_f32(
                    false, a, false, b, (short)0, acc[nt], false, false);
            }
        }
        __syncthreads();                 // all waves done with buffer `cur`
    }
#else
    // Fallback: cooperative global->LDS staging (single buffer)
    const int arow  = tid >> 1;
    const int akoff = (tid & 1) * 8;
    const int bcol  = tid >> 2;
    const int bkoff = (tid & 3) * 4;
    for (int i = 0; i < nIter; ++i) {
        const int k0 = i * BK_;
        float4 a0 = *(const float4*)(Abase + (size_t)arow * GK_ + k0 + akoff);
        float4 a1 = *(const float4*)(Abase + (size_t)arow * GK_ + k0 + akoff + 4);
        *(float4*)(&As[0][arow * BK_ + akoff])     = a0;
        *(float4*)(&As[0][arow * BK_ + akoff + 4]) = a1;
        float4 bv = *(const float4*)(Bbase + (size_t)bcol * GK_ + k0 + bkoff);
        *(float4*)(&Bs[0][bcol * BK_ + bkoff]) = bv;
        __syncthreads();
#pragma unroll
        for (int kk = 0; kk < BK_; kk += 4) {
            v2f a = *(const v2f*)(&As[0][(wave * 16 + l16) * BK_ + kk + 2 * g]);
#pragma unroll
            for (int nt = 0; nt < 4; ++nt) {
                v2f b = *(const v2f*)(&Bs[0][(nt * 16 + l16) * BK_ + kk + 2 * g]);
                acc[nt] = __builtin_amdgcn_wmma_f32_16x16x4_f32(
                    false, a, false, b, (short)0, acc[nt], false, false);
            }
        }
        __syncthreads();
    }
#endif

    // ---- epilogue: C/D layout: VGPR r -> row r + 8*g, lane%16 -> col ----
#pragma unroll
    for (int nt = 0; nt < 4; ++nt) {
#pragma unroll
        for (int r = 0; r < 8; ++r) {
            int rowT = wave * 16 + r + 8 * g;
            int col  = tileN + nt * 16 + l16;
            int R    = tileM + rowT;
            float val = (acc[nt][r] + bias[col]) * scale;
            if (layout == 0) {
                int n = R / B_, b = R % B_;      // input rows ordered (n, b)
                int h = col >> 6, e = col & 63;
                OUT[(((size_t)b * H_ + h) * KN_ + n) * E_ + e] = val;
            } else {
                OUT[(size_t)R * GN_ + col] = val;
            }
        }
    }
}

// =====================================================================
// Metric: M[bh][n] = max_s(q.k_samp) - sum_s(q.k_samp)/KN
// =====================================================================
__global__ __launch_bounds__(256)
void metric_kernel(const float* __restrict__ Q, const float* __restrict__ Kd,
                   const int* __restrict__ idxs, float* __restrict__ Mout)
{
    int t = blockIdx.x * 256 + threadIdx.x;
    if (t >= B_ * H_ * KN_) return;
    int n  = t % KN_;
    int bh = t / KN_;
    const float* q = Q + ((size_t)bh * KN_ + n) * E_;
    float4 qr[E_ / 4];
#pragma unroll
    for (int e = 0; e < E_ / 4; ++e) qr[e] = *(const float4*)(q + 4 * e);

    float mx = -3.0e38f, sm = 0.0f;
    for (int s = 0; s < U_; ++s) {
        int j = idxs[n * U_ + s];
        const float* kp = Kd + ((size_t)bh * KN_ + j) * E_;
        float d = 0.0f;
#pragma unroll
        for (int e = 0; e < E_ / 4; ++e) {
            float4 kv = *(const float4*)(kp + 4 * e);
            d += qr[e].x * kv.x + qr[e].y * kv.y + qr[e].z * kv.z + qr[e].w * kv.w;
        }
        mx = fmaxf(mx, d);
        sm += d;
    }
    Mout[t] = mx - sm * (1.0f / KN_);
}

// =====================================================================
// Top-U_ per (b,h): iterative argmax over 2048 values in LDS
// =====================================================================
__global__ __launch_bounds__(256)
void topk_kernel(const float* __restrict__ Mm, int* __restrict__ Mtop)
{
    __shared__ float vals[KN_];
    __shared__ float rmax[256];
    __shared__ int   ridx[256];
    int bh = blockIdx.x, t = threadIdx.x;
    for (int i = t; i < KN_; i += 256) vals[i] = Mm[(size_t)bh * KN_ + i];
    __syncthreads();

    for (int it = 0; it < U_; ++it) {
        float bm = -3.0e38f; int bi = 0x7fffffff;
        for (int i = t; i < KN_; i += 256) {
            float v = vals[i];
            if (v > bm || (v == bm && i < bi)) { bm = v; bi = i; }
        }
        rmax[t] = bm; ridx[t] = bi;
        __syncthreads();
        for (int off = 128; off > 0; off >>= 1) {
            if (t < off) {
                float v2 = rmax[t + off]; int i2 = ridx[t + off];
                if (v2 > rmax[t] || (v2 == rmax[t] && i2 < ridx[t])) { rmax[t] = v2; ridx[t] = i2; }
            }
            __syncthreads();
        }
        if (t == 0) { Mtop[bh * U_ + it] = ridx[0]; vals[ridx[0]] = -3.0e38f; }
        __syncthreads();
    }
}

// =====================================================================
// Attention for the 40 selected queries: one block per (b,h,s)
// =====================================================================
__global__ __launch_bounds__(256)
void attn_kernel(const float* __restrict__ Q, const float* __restrict__ Kd,
                 const float* __restrict__ V, const int* __restrict__ Mtop,
                 float* __restrict__ upd)
{
    __shared__ float sc[KN_];
    __shared__ float qs[E_];
    __shared__ float red[256];
    int blk = blockIdx.x;
    int s   = blk % U_;
    int bh  = blk / U_;
    int t   = threadIdx.x;
    int nq  = Mtop[bh * U_ + s];
    if (t < E_) qs[t] = Q[((size_t)bh * KN_ + nq) * E_ + t];
    __syncthreads();

    float lm = -3.0e38f;
    for (int n = t; n < KN_; n += 256) {
        const float* kp = Kd + ((size_t)bh * KN_ + n) * E_;
        float d = 0.0f;
#pragma unroll
        for (int e = 0; e < E_; e += 4) {
            float4 kv = *(const float4*)(kp + e);
            d += qs[e] * kv.x + qs[e + 1] * kv.y + qs[e + 2] * kv.z + qs[e + 3] * kv.w;
        }
        sc[n] = d;
        lm = fmaxf(lm, d);
    }
    red[t] = lm; __syncthreads();
    for (int off = 128; off > 0; off >>= 1) {
        if (t < off) red[t] = fmaxf(red[t], red[t + off]);
        __syncthreads();
    }
    float gm = red[0]; __syncthreads();

    float ls = 0.0f;
    for (int n = t; n < KN_; n += 256) {
        float ev = expf(sc[n] - gm);
        sc[n] = ev;
        ls += ev;
    }
    red[t] = ls; __syncthreads();
    for (int off = 128; off > 0; off >>= 1) {
        if (t < off) red[t] += red[t + off];
        __syncthreads();
    }
    float inv = 1.0f / red[0]; __syncthreads();

    int e = t & 63, part = t >> 6;
    float acc = 0.0f;
    for (int n = part; n < KN_; n += 4)
        acc += sc[n] * V[((size_t)bh * KN_ + n) * E_ + e];
    red[t] = acc; __syncthreads();
    if (part == 0) {
        float r = red[t] + red[t + 64] + red[t + 128] + red[t + 192];
        upd[((size_t)bh * U_ + s) * E_ + e] = r * inv;
    }
}

// =====================================================================
// V mean over n per (b,h,e)
// =====================================================================
__global__ __launch_bounds__(256)
void vmean_kernel(const float* __restrict__ V, float* __restrict__ vmean)
{
    __shared__ float red[256];
    int bh = blockIdx.x, t = threadIdx.x;
    int e = t & 63, part = t >> 6;
    float acc = 0.0f;
    for (int n = part; n < KN_; n += 4)
        acc += V[((size_t)bh * KN_ + n) * E_ + e];
    red[t] = acc; __syncthreads();
    if (part == 0)
        vmean[bh * E_ + e] = (red[t] + red[t + 64] + red[t + 128] + red[t + 192]) * (1.0f / KN_);
}

// ctx[b][n][c] = vmean[b][c]   (c = h*64+e; vmean laid [b*512 + c])
__global__ __launch_bounds__(256)
void ctx_fill_kernel(const float* __restrict__ vmean, float* __restrict__ ctx)
{
    int i = blockIdx.x * 256 + threadIdx.x;      // over B_*KN_*D_
    int c = i % D_;
    int b = (i / D_) / KN_;
    ctx[i] = vmean[b * D_ + c];
}

// scatter the 40 updated rows per (b,h)
__global__ __launch_bounds__(256)
void ctx_scatter_kernel(const float* __restrict__ upd, const int* __restrict__ Mtop,
                        float* __restrict__ ctx)
{
    int i = blockIdx.x * 256 + threadIdx.x;      // over B_*H_*U_*E_
    int e  = i & 63;
    int su = (i >> 6) % U_;
    int bh = (i >> 6) / U_;
    int b = bh / H_, h = bh % H_;
    int n = Mtop[bh * U_ + su];
    ctx[((size_t)b * KN_ + n) * D_ + h * E_ + e] = upd[i];
}

// =====================================================================
extern "C" void kernel_launch(void* const* d_in, const int* in_sizes, int n_in,
                              void* d_out, int out_size, void* d_ws, size_t ws_size,
                              hipStream_t stream)
{
    const float* query = (const float*)d_in[0];
    const float* key   = (const float*)d_in[1];
    const float* value = (const float*)d_in[2];
    const int*   idxs  = (const int*)d_in[3];
    const float* Wq = (const float*)d_in[4];  const float* bq = (const float*)d_in[5];
    const float* Wk = (const float*)d_in[6];  const float* bk = (const float*)d_in[7];
    const float* Wv = (const float*)d_in[8];  const float* bv = (const float*)d_in[9];
    const float* Wo = (const float*)d_in[10]; const float* bo = (const float*)d_in[11];
    float* out = (float*)d_out;

    // workspace layout (bytes)
    char* ws = (char*)d_ws;
    size_t off = 0;
    const size_t bhne = (size_t)B_ * H_ * KN_ * E_;          // 4,194,304 floats
    float* Qp    = (float*)(ws + off); off += bhne * 4;
    float* Kp    = (float*)(ws + off); off += bhne * 4;
    float* Vp    = (float*)(ws + off); off += bhne * 4;
    float* ctx   = (float*)(ws + off); off += (size_t)B_ * KN_ * D_ * 4;
    float* Mm    = (float*)(ws + off); off += (size_t)B_ * H_ * KN_ * 4;
    float* upd   = (float*)(ws + off); off += (size_t)B_ * H_ * U_ * E_ * 4;
    float* vmean = (float*)(ws + off); off += (size_t)B_ * H_ * E_ * 4;
    int*   Mtop  = (int*)(ws + off);   off += (size_t)B_ * H_ * U_ * 4;

    const float scaling = 0.125f;    // E^-0.5 = 1/8

    dim3 gg(GN_ / BN_, GM_ / BM_);   // (8, 64)

    // q/k/v projections (WMMA + TDM staging)
    wmma_gemm_kernel<<<gg, 256, 0, stream>>>(query, Wq, bq, scaling, Qp, 0);
    wmma_gemm_kernel<<<gg, 256, 0, stream>>>(key,   Wk, bk, 1.0f,    Kp, 0);
    wmma_gemm_kernel<<<gg, 256, 0, stream>>>(value, Wv, bv, 1.0f,    Vp, 0);

    // sparse metric + top-k selection
    metric_kernel<<<(B_ * H_ * KN_) / 256, 256, 0, stream>>>(Qp, Kp, idxs, Mm);
    topk_kernel<<<B_ * H_, 256, 0, stream>>>(Mm, Mtop);

    // attention on selected queries
    attn_kernel<<<B_ * H_ * U_, 256, 0, stream>>>(Qp, Kp, Vp, Mtop, upd);

    // context assembly
    vmean_kernel<<<B_ * H_, 256, 0, stream>>>(Vp, vmean);
    ctx_fill_kernel<<<(B_ * KN_ * D_) / 256, 256, 0, stream>>>(vmean, ctx);
    ctx_scatter_kernel<<<(B_ * H_ * U_ * E_) / 256, 256, 0, stream>>>(upd, Mtop, ctx);

    // output projection
    wmma_gemm_kernel<<<gg, 256, 0, stream>>>(ctx, Wo, bo, 1.0f, out, 1);
}